// UnbalancedSCOTHead_52037823758977
// MI455X (gfx1250) — compile-verified
//
#include <hip/hip_runtime.h>
#include <math.h>

// ---------------------------------------------------------------------------
// UnbalancedSCOTHead for MI455X (gfx1250, wave32).
// Exploits the type-mask structure: the 8192x1088 Sinkhorn decomposes into 17
// independent ~482x64 sub-problems (masked entries underflow to exact 0 in
// f32), so the 50-iteration loop runs in ONE kernel with workgroup barriers
// only, and each sub-problem's logK matrix is staged once into CDNA5's 320KB
// LDS (stride-65 padding -> bank-conflict-free row AND column passes).
// GEMMs use V_WMMA_F32_16X16X4_F32 (full f32 precision; d^2 is scaled by
// 1/tau=20 before exp, so bf16 inputs would be numerically unacceptable).
// ---------------------------------------------------------------------------

typedef __attribute__((ext_vector_type(2))) float v2f;
typedef __attribute__((ext_vector_type(8))) float v8f;

#define NJT 17
#define INV_TAU 20.0f               // 1 / 0.05
#define KLF (1.0f / 1.05f)          // rho / (rho + tau)
#define BIGC 1.0e6f
#define SINK_ITERS 50
#define MAX_ROWS 2048               // >> 74 sigma above E[N/17]=482
#define CACHE_ROWS 1024             // logK rows staged in LDS (>25 sigma)
#define CACHE_STRIDE 65             // 64 + 1 pad: conflict-free column pass
#define SELU_L 1.0507009873554805f
#define SELU_A 1.6732632423543772f

__device__ __forceinline__ v8f wmma_f32x4(v2f a, v2f b, v8f c) {
  // D = A(16x4,f32) * B(4x16,f32) + C(16x16,f32)
  return __builtin_amdgcn_wmma_f32_16x16x4_f32(
      /*neg_a=*/false, a, /*neg_b=*/false, b,
      /*c_mod=*/(short)0, c, /*reuse_a=*/false, /*reuse_b=*/false);
}

// --------------------------- ||prototype||^2 -------------------------------
__global__ __launch_bounds__(32) void k_psq(const float* __restrict__ proto,
                                            float* __restrict__ psq, int H) {
  const int p = blockIdx.x, lane = threadIdx.x;
  float s = 0.f;
  for (int c = lane; c < H; c += 32) { float v = proto[(size_t)p * H + c]; s += v * v; }
  for (int d = 16; d; d >>= 1) s += __shfl_xor(s, d, 32);
  if (lane == 0) psq[p] = s;
}

// ------- fused: h = selu(emb@W1+b1); logK = -max(d2(h,proto),0)/tau --------
// One workgroup (8 waves) per 16-row block. h tile lives entirely in LDS.
__global__ __launch_bounds__(256) void k_encode(
    const float* __restrict__ emb, const float* __restrict__ W1,
    const float* __restrict__ b1, const float* __restrict__ proto,
    const float* __restrict__ psq, float* __restrict__ logK,
    int D, int H, int K) {
  __shared__ float hTile[16][516];        // stride 516: conflict-free WMMA reads
  __shared__ float hsq_s[16];
  __shared__ float part[4][16][16];       // K-split partial dot products

  const int wv   = threadIdx.x >> 5;
  const int lane = threadIdx.x & 31;
  const int m    = lane & 15;             // A: row / B,C,D: column-in-tile
  const int ko   = (lane >> 4) << 1;      // A/B K sub-offset {0,2}
  const int row0 = blockIdx.x * 16;

  // Phase 1: 16 x H tile of h, each wave owns 16 columns per 128-col sweep.
  for (int cb = 0; cb < H; cb += 128) {
    const int n0 = cb + wv * 16;
    v8f acc = {};
    for (int kk = 0; kk < D; kk += 4) {
      v2f a, b;
      const float* ap = emb + (size_t)(row0 + m) * D + kk + ko;
      a.x = ap[0]; a.y = ap[1];                       // A[m, kk+ko+{0,1}]
      const float* bp = W1 + (size_t)(kk + ko) * H + n0 + m;
      b.x = bp[0]; b.y = bp[H];                       // B[kk+ko+{0,1}, m]
      __builtin_prefetch(bp + (size_t)8 * H, 0, 3);   // near-cache prefetch
      acc = wmma_f32x4(a, b, acc);
    }
    const float bias = b1[n0 + m];
    for (int r = 0; r < 8; ++r) {
      float x = acc[r] + bias;                        // C/D: M = r + 8*(lane>>4)
      x = (x > 0.f) ? (SELU_L * x) : (SELU_L * SELU_A * expm1f(x));
      hTile[r + ((lane >> 4) << 3)][n0 + m] = x;
    }
  }
  __syncthreads();

  // Row sums of h^2: wave wv handles rows 2wv, 2wv+1 (16 lanes per row).
  {
    const int r = (wv << 1) + (lane >> 4);
    float s = 0.f;
    for (int c = m; c < H; c += 16) { float v = hTile[r][c]; s += v * v; }
    for (int d = 8; d; d >>= 1) s += __shfl_xor(s, d, 32);
    if (m == 0) hsq_s[r] = s;
  }
  __syncthreads();

  // Phase 2: dot = h @ proto^T. pt = prototype tile (0..3), kh = K half.
  const int pt = wv & 3, kh = wv >> 2;
  const int kb = kh * (H >> 1);
  v8f acc = {};
  for (int kk = kb; kk < kb + (H >> 1); kk += 4) {
    v2f a, b;
    a.x = hTile[m][kk + ko]; a.y = hTile[m][kk + ko + 1];
    const float* bp = proto + (size_t)(pt * 16 + m) * H + kk + ko;
    b.x = bp[0]; b.y = bp[1];
    acc = wmma_f32x4(a, b, acc);
  }
  if (kh == 1)
    for (int r = 0; r < 8; ++r) part[pt][r + ((lane >> 4) << 3)][m] = acc[r];
  __syncthreads();
  if (kh == 0) {
    for (int r = 0; r < 8; ++r) {
      const int row = r + ((lane >> 4) << 3);
      const int p = pt * 16 + m;
      const float dot = acc[r] + part[pt][row][m];
      float d2 = hsq_s[row] + psq[p] - 2.f * dot;
      d2 = fmaxf(d2, 0.f);
      logK[(size_t)(row0 + row) * K + p] = -d2 * INV_TAU;
    }
  }
}

// ----------------- per-type unbalanced Sinkhorn (50 iters) -----------------
// 17 workgroups; each owns one joint type's independent [cnt x 64] problem.
// logK rows staged into dynamic LDS (CDNA5: up to 320KB/WGP) with stride 65.
__global__ __launch_bounds__(1024) void k_sinkhorn(
    const float* __restrict__ logK, const int* __restrict__ jt,
    float* __restrict__ log_u_g, float* __restrict__ log_v_g, int N, int K) {
  extern __shared__ float ldsK[];                    // CACHE_ROWS * 65 floats
  const int t = blockIdx.x;
  __shared__ int   rows[MAX_ROWS];
  __shared__ float lu[MAX_ROWS];
  __shared__ float lv[64];
  __shared__ int   wpre[32];
  __shared__ int   base_s;

  const int wv = threadIdx.x >> 5, lane = threadIdx.x & 31;
  if (threadIdx.x < 64) lv[threadIdx.x] = 0.f;       // log_v init = 0
  if (threadIdx.x == 0) base_s = 0;
  __syncthreads();

  // Deterministic compaction (ballot + scan): row order is always ascending n,
  // so reductions are bit-reproducible across graph replays.
  for (int base = 0; base < N; base += 1024) {
    const int n = base + threadIdx.x;
    const bool match = (n < N) && (jt[n] == t);
    const unsigned mask = (unsigned)__ballot(match);
    if (lane == 0) wpre[wv] = __popc(mask);
    __syncthreads();
    if (wv == 0) {
      int v = wpre[lane];
      for (int d = 1; d < 32; d <<= 1) { int o = __shfl_up(v, d, 32); if (lane >= d) v += o; }
      wpre[lane] = v;                                // inclusive scan
    }
    __syncthreads();
    if (match) {
      const int pos = base_s + (wv ? wpre[wv - 1] : 0) + __popc(mask & ((1u << lane) - 1));
      if (pos < MAX_ROWS) rows[pos] = n;
    }
    __syncthreads();
    if (threadIdx.x == 0) base_s += wpre[31];
    __syncthreads();
  }
  const int cnt  = min(base_s, MAX_ROWS);
  const int ccnt = min(cnt, CACHE_ROWS);

  // Stage this type's logK rows into LDS (coalesced: consecutive threads read
  // consecutive columns of one row).
  for (int e = threadIdx.x; e < ccnt * 64; e += 1024) {
    const int i = e >> 6, c = e & 63;
    ldsK[i * CACHE_STRIDE + c] = logK[(size_t)rows[i] * K + c];
  }
  __syncthreads();

  for (int it = 0; it < SINK_ITERS; ++it) {
    // Row update: one wave per row; 64 slots split as (lane, lane+32).
    for (int i = wv; i < cnt; i += 32) {
      float lk0, lk1;
      if (i < CACHE_ROWS) {
        lk0 = ldsK[i * CACHE_STRIDE + lane];
        lk1 = ldsK[i * CACHE_STRIDE + lane + 32];
      } else {                                       // statistically unreachable
        const float* g = logK + (size_t)rows[i] * K;
        lk0 = g[lane]; lk1 = g[lane + 32];
      }
      const float x0 = lk0 + lv[lane];
      const float x1 = lk1 + lv[lane + 32];
      float mx = fmaxf(x0, x1);
      for (int d = 16; d; d >>= 1) mx = fmaxf(mx, __shfl_xor(mx, d, 32));
      float s = __expf(x0 - mx) + __expf(x1 - mx);
      for (int d = 16; d; d >>= 1) s += __shfl_xor(s, d, 32);
      if (lane == 0) lu[i] = -(mx + __logf(s));
    }
    __syncthreads();
    // Column update: wave wv owns columns wv and wv+32; online logsumexp.
    // LDS banks hit are (i + c) mod 64 -> conflict-free across lane-strided i.
    for (int cc = 0; cc < 2; ++cc) {
      const int c = wv + 32 * cc;
      float mx = -3.0e38f, s = 0.f;
      for (int i = lane; i < cnt; i += 32) {
        const float lk = (i < CACHE_ROWS) ? ldsK[i * CACHE_STRIDE + c]
                                          : logK[(size_t)rows[i] * K + c];
        const float x = lk + lu[i];
        const float mn = fmaxf(mx, x);
        s = s * __expf(mx - mn) + __expf(x - mn);
        mx = mn;
      }
      for (int d = 16; d; d >>= 1) {
        const float mo = __shfl_xor(mx, d, 32), so = __shfl_xor(s, d, 32);
        const float mn = fmaxf(mx, mo);
        s = s * __expf(mx - mn) + so * __expf(mo - mn);
        mx = mn;
      }
      if (lane == 0) {
        const float log_s = (s > 0.f) ? (mx + __logf(s)) : (-BIGC * INV_TAU);
        lv[c] = KLF * (0.0f - log_s);               // log_c == 0
      }
    }
    __syncthreads();
  }

  for (int i = threadIdx.x; i < cnt; i += 1024) log_u_g[rows[i]] = lu[i];
  if (threadIdx.x < 64) log_v_g[threadIdx.x * NJT + t] = lv[threadIdx.x];
}

// ----------------------- T + logits materialization ------------------------
__global__ __launch_bounds__(256) void k_finalize(
    const float* __restrict__ logK, const int* __restrict__ jt,
    const float* __restrict__ log_u, const float* __restrict__ log_v,
    float* __restrict__ out, int N, int K) {
  const int S = K * NJT;
  const int gid = blockIdx.x * 256 + threadIdx.x;
  if (gid >= N * S) return;
  const int n = gid / S, s = gid - n * S;
  const int k = s / NJT, t = s - k * NJT;
  float* logits = out;
  float* T = out + (size_t)N * K;
  float v = 0.f;                                   // masked: exp underflows to 0
  if (jt[n] == t) {
    v = __expf(logK[(size_t)n * K + k] + log_u[n] + log_v[s]);
    logits[(size_t)n * K + k] = __logf(v + 1e-8f); // person_mass has 1 nonzero slot
  }
  T[gid] = v;
}

// ---------------------------------------------------------------------------
extern "C" void kernel_launch(void* const* d_in, const int* in_sizes, int n_in,
                              void* d_out, int out_size, void* d_ws, size_t ws_size,
                              hipStream_t stream) {
  const float* emb   = (const float*)d_in[0];
  const int*   jt    = (const int*)d_in[1];
  const float* W1    = (const float*)d_in[2];
  const float* b1    = (const float*)d_in[3];
  const float* proto = (const float*)d_in[4];

  const int N = in_sizes[1];
  const int D = in_sizes[0] / N;       // 256
  const int H = in_sizes[3];           // 512
  const int K = in_sizes[4] / H;       // 64

  float* wsf   = (float*)d_ws;
  float* logK  = wsf;                               // N*K        (2 MB)
  float* log_u = logK + (size_t)N * K;              // N
  float* log_v = log_u + N;                         // K*17
  float* psq   = log_v + (size_t)K * NJT;           // K

  hipLaunchKernelGGL(k_psq, dim3(K), dim3(32), 0, stream, proto, psq, H);
  hipLaunchKernelGGL(k_encode, dim3(N / 16), dim3(256), 0, stream,
                     emb, W1, b1, proto, psq, logK, D, H, K);
  const size_t shmem = (size_t)CACHE_ROWS * CACHE_STRIDE * sizeof(float); // 260 KB
  hipLaunchKernelGGL(k_sinkhorn, dim3(NJT), dim3(1024), shmem, stream,
                     logK, jt, log_u, log_v, N, K);
  const int total = N * K * NJT;
  hipLaunchKernelGGL(k_finalize, dim3((total + 255) / 256), dim3(256), 0, stream,
                     logK, jt, log_u, log_v, (float*)d_out, N, K);
}